// YoloWrapper_65481071395015
// MI455X (gfx1250) — compile-verified
//
#include <hip/hip_runtime.h>
#include <stdint.h>

typedef float v2f __attribute__((ext_vector_type(2)));
typedef float v8f __attribute__((ext_vector_type(8)));

#define CONF_TH  0.25f
#define IOU_TH   0.45f
#define EPS_IOU  1e-7f
#define KNMS     1000
#define KPAD     1024
#define MAXDET   300
#define NSEL     2048
#define NBKT     2048
#define BASE2    0x3E7FE000u   // float-bits(0.25) - 0x2000: conf>0.25  =>  bucket >= 1

// ---- dynamic LDS layout (bytes) ----
#define OFF_KEYS   0u        // u64[2048]   16384 B
#define OFF_HIST   16384u    // u32[2048]    8192 B
#define OFF_SUF    24576u    // u32[2048]    8192 B
#define OFF_BOX4   32768u    // float4[1024] 16384 B (x1,y1,x2,y2 packed)
#define OFF_BAR    49152u    // f32[1024] areas
#define OFF_BSC    53248u    // f32[1024] scores
#define OFF_SUP    57344u    // u32[1024*32] 131072 B suppression bit-matrix
#define OFF_KEEP   188416u   // u32[32]
#define OFF_CNT    188544u   // u32[32]
#define OFF_OIDX   188672u   // int[300]
#define OFF_SCAL   189872u   // u32[4]: 0=selCount 1=threshBucket
#define SMEM_BYTES 189952u

#if __has_builtin(__builtin_amdgcn_ballot_w32)
#define BALLOT32(p) __builtin_amdgcn_ballot_w32(p)
#else
#define BALLOT32(p) ((unsigned)__ballot(p))
#endif

// IEEE f32 subtract that the FMA-contraction pass cannot fuse (matches the
// reference's separate mul-then-sub rounding).
__device__ __forceinline__ float fsub_nc(float a, float b) {
    float r;
    asm("v_sub_f32 %0, %1, %2" : "=v"(r) : "v"(a), "v"(b));
    return r;
}

__global__ __launch_bounds__(1024) void yolo_nms_kernel(
    const float* __restrict__ pred, float* __restrict__ out, int N)
{
    extern __shared__ char smem[];
    unsigned long long* keys = (unsigned long long*)(smem + OFF_KEYS);
    unsigned* hist = (unsigned*)(smem + OFF_HIST);
    unsigned* suf  = (unsigned*)(smem + OFF_SUF);
    float4*   box4 = (float4*)(smem + OFF_BOX4);
    float*    bar  = (float*)(smem + OFF_BAR);
    float*    bsc  = (float*)(smem + OFF_BSC);
    unsigned* supmat   = (unsigned*)(smem + OFF_SUP);
    unsigned* keepMask = (unsigned*)(smem + OFF_KEEP);
    unsigned* cnts     = (unsigned*)(smem + OFF_CNT);
    int*      outIdx   = (int*)(smem + OFF_OIDX);
    unsigned* scal     = (unsigned*)(smem + OFF_SCAL);

    const int tid  = threadIdx.x;
    const int wave = tid >> 5;
    const int lane = tid & 31;
    const int b    = blockIdx.x;
    const float* p = pred + (size_t)b * (size_t)N * 6u;

    // Exact emulation of "RN(inter/denom) > 0.45f" without a division:
    // t = 0.45f (0x3EE66666, even mantissa LSB => midpoint ties round back to t),
    // so RN(r) > t  <=>  r > mid(t, succ(t)).  MID and MID*(f64)denom are exact
    // in double (25b x 24b significands), hence the predicate is bit-exact.
    const float  T0  = __uint_as_float(0x3EE66666u);
    const float  T1  = __uint_as_float(0x3EE66667u);
    const double MID = ((double)T0 + (double)T1) * 0.5;

    // ---------------- Phase 1: score histogram ----------------
    for (int i = tid; i < NBKT; i += 1024) hist[i] = 0u;
    if (tid == 0) { scal[0] = 0u; scal[1] = 0u; }
    __syncthreads();

    for (int n = tid; n < N; n += 1024) {
        float conf = p[(size_t)n * 6 + 4] * p[(size_t)n * 6 + 5];
        unsigned u = __float_as_uint(conf);
        int bkt = (u <= BASE2) ? 0 : (int)((u - BASE2) >> 13);
        if (bkt > NBKT - 1) bkt = NBKT - 1;
        atomicAdd(&hist[bkt], 1u);
    }
    __syncthreads();

    // inclusive suffix sums over buckets (Hillis-Steele, double-buffered)
    unsigned* src = hist; unsigned* dst = suf;
    for (int off = 1; off < NBKT; off <<= 1) {
        for (int i = tid; i < NBKT; i += 1024)
            dst[i] = src[i] + ((i + off < NBKT) ? src[i + off] : 0u);
        __syncthreads();
        unsigned* t = src; src = dst; dst = t;
    }
    for (int i = tid; i < NBKT; i += 1024)
        if (src[i] >= (unsigned)KNMS) atomicMax(&scal[1], (unsigned)i);
    __syncthreads();
    int Tsel = (int)scal[1]; if (Tsel < 1) Tsel = 1;

    // ---------------- Phase 2: compact candidates ----------------
    for (int n = tid; n < N; n += 1024) {
        float conf = p[(size_t)n * 6 + 4] * p[(size_t)n * 6 + 5];
        if (conf > CONF_TH) {
            unsigned u = __float_as_uint(conf);
            int bkt = (int)((u - BASE2) >> 13);
            if (bkt > NBKT - 1) bkt = NBKT - 1;
            if (bkt >= Tsel) {
                unsigned pos = atomicAdd(&scal[0], 1u);
                if (pos < (unsigned)NSEL)
                    keys[pos] = ((unsigned long long)u << 32) |
                                (unsigned long long)(0xFFFFFFFFu - (unsigned)n);
            }
        }
    }
    __syncthreads();
    unsigned selCount = scal[0]; if (selCount > NSEL) selCount = NSEL;
    int validCnt = (selCount > (unsigned)KNMS) ? KNMS : (int)selCount;
    for (int i = tid; i < NSEL; i += 1024)
        if ((unsigned)i >= selCount) keys[i] = 0ull;

    // ---------------- Phase 3: bitonic sort, descending ----------------
    for (unsigned k = 2; k <= (unsigned)NSEL; k <<= 1) {
        for (unsigned j = k >> 1; j > 0; j >>= 1) {
            __syncthreads();
            for (unsigned idx = tid; idx < (unsigned)NSEL; idx += 1024) {
                unsigned ixj = idx ^ j;
                if (ixj > idx) {
                    unsigned long long a = keys[idx], c = keys[ixj];
                    bool up = ((idx & k) == 0);
                    if (up ? (a < c) : (a > c)) { keys[idx] = c; keys[ixj] = a; }
                }
            }
        }
    }
    __syncthreads();

    // ---------------- Phase 4: build sorted box rows ----------------
    {
        unsigned long long kk = keys[tid];
        float s = __uint_as_float((unsigned)(kk >> 32));
        bool valid = (tid < KNMS) && (s > CONF_TH);
        float x1 = 0.f, y1 = 0.f, x2 = 0.f, y2 = 0.f, area = 0.f;
        if (valid) {
            unsigned n = 0xFFFFFFFFu - (unsigned)(kk & 0xFFFFFFFFull);
            const float* q = p + (size_t)n * 6;
            float cx = q[0], cy = q[1], w = q[2], h = q[3];
            x1 = cx - w * 0.5f; y1 = cy - h * 0.5f;   // *0.5f is exact, no contraction issue
            x2 = cx + w * 0.5f; y2 = cy + h * 0.5f;
            area = (x2 - x1) * (y2 - y1);
        }
        box4[tid] = make_float4(x1, y1, x2, y2);      // one ds_store_b128
        bar[tid] = area; bsc[tid] = s;
        unsigned vb = BALLOT32(valid);
        if (lane == 0) keepMask[wave] = vb;
    }
    __syncthreads();

    // ---------------- Phase 5: suppression bit-matrix via WMMA ----------------
    // Tiles are 16 rows x 32 cols; areaSum tile = [area_m,1] x [1;area_n]^T on the matrix unit.
    for (int t = wave; t < (KPAD / 16) * 32; t += 32) {
        int rT = t >> 5;            // row tile
        int dn = t & 31;            // dword (32-col block)
        int m0 = rT << 4;
        int n0 = dn << 5;
        if (dn * 32 + 31 < m0) continue;            // lower triangle never read (masked j<=i)
        if (m0 >= validCnt || n0 >= validCnt) continue;  // rows/cols with keep-bit==0 never matter

        v2f A; A[0] = (lane < 16) ? bar[m0 + (lane & 15)] : 0.0f;
               A[1] = (lane < 16) ? 1.0f : 0.0f;
        v2f B0; B0[0] = (lane < 16) ? 1.0f : 0.0f;
                B0[1] = (lane < 16) ? bar[n0 + (lane & 15)] : 0.0f;
        v2f B1; B1[0] = B0[0];
                B1[1] = (lane < 16) ? bar[n0 + 16 + (lane & 15)] : 0.0f;
        v8f C = {0.f,0.f,0.f,0.f,0.f,0.f,0.f,0.f};
        v8f D0 = __builtin_amdgcn_wmma_f32_16x16x4_f32(false, A, false, B0, (short)0, C, false, false);
        v8f D1 = __builtin_amdgcn_wmma_f32_16x16x4_f32(false, A, false, B1, (short)0, C, false, false);

        int nA = n0 + (lane & 15);
        float4 nb0 = box4[nA];          // one ds_load_b128 per column block
        float4 nb1 = box4[nA + 16];
        int mbase = m0 + ((lane >= 16) ? 8 : 0);

#pragma unroll
        for (int v = 0; v < 8; ++v) {
            float4 mb = box4[mbase + v];   // one ds_load_b128 per row
            // block 0
            float iw0 = fmaxf(fminf(mb.z, nb0.z) - fmaxf(mb.x, nb0.x), 0.0f);
            float ih0 = fmaxf(fminf(mb.w, nb0.w) - fmaxf(mb.y, nb0.y), 0.0f);
            float in0 = iw0 * ih0;
            float d0  = fsub_nc(D0[v], in0) + EPS_IOU;   // un-contractible sub
            bool bit0 = (double)in0 > MID * (double)d0;
            // block 1
            float iw1 = fmaxf(fminf(mb.z, nb1.z) - fmaxf(mb.x, nb1.x), 0.0f);
            float ih1 = fmaxf(fminf(mb.w, nb1.w) - fmaxf(mb.y, nb1.y), 0.0f);
            float in1 = iw1 * ih1;
            float d1  = fsub_nc(D1[v], in1) + EPS_IOU;
            bool bit1 = (double)in1 > MID * (double)d1;

            unsigned q0 = BALLOT32(bit0);
            unsigned q1 = BALLOT32(bit1);
            if (lane == 0) supmat[(m0 + v) * 32 + dn]     = (q0 & 0xFFFFu) | ((q1 & 0xFFFFu) << 16);
            if (lane == 1) supmat[(m0 + v + 8) * 32 + dn] = (q0 >> 16)     | (q1 & 0xFFFF0000u);
        }
    }
    __syncthreads();

    // ---------------- Phase 6: greedy reduce (wave 0, register keep-bits) ----------------
    if (wave == 0) {
        unsigned kreg = keepMask[lane];
        int base = lane << 5;
        int i = -1;
        while (true) {
            unsigned m = kreg;
            if (i >= base + 31)      m = 0u;
            else if (i >= base)      m &= ~(unsigned)((2ull << (i - base)) - 1ull);
            int cand = m ? (base + __builtin_ctz(m)) : 0x7FFFFFFF;
            for (int off = 16; off; off >>= 1) {
                int o = __shfl_xor(cand, off, 32);
                cand = (o < cand) ? o : cand;
            }
            if (cand >= KNMS) break;
            i = cand;
            unsigned sm = supmat[i * 32 + lane];
            if (i >= base + 31)      sm = 0u;
            else if (i >= base)      sm &= ~(unsigned)((2ull << (i - base)) - 1ull);
            kreg &= ~sm;
        }
        keepMask[lane] = kreg;
    }
    __syncthreads();

    // ---------------- Phase 7: rank + write output ----------------
    if (tid < 32)     cnts[tid] = __popc(keepMask[tid]);
    if (tid < MAXDET) outIdx[tid] = -1;
    __syncthreads();
    if (tid == 0) {
        unsigned run = 0;
        for (int w = 0; w < 32; ++w) { unsigned c = cnts[w]; cnts[w] = run; run += c; }
    }
    __syncthreads();
    {
        unsigned mw = keepMask[tid >> 5];
        if ((mw >> (tid & 31)) & 1u) {
            unsigned low = mw & ((1u << (tid & 31)) - 1u);
            int rank = (int)cnts[tid >> 5] + __popc(low);
            if (rank < MAXDET) outIdx[rank] = tid;
        }
    }
    __syncthreads();
    if (tid < MAXDET) {
        float o0 = 0.f, o1 = 0.f, o2 = 0.f, o3 = 0.f, o4 = 0.f;
        int s = outIdx[tid];
        if (s >= 0) {
            float4 bb = box4[s];
            o0 = bb.x; o1 = bb.y; o2 = bb.z; o3 = bb.w; o4 = bsc[s];
        }
        float* orow = out + ((size_t)b * MAXDET + (size_t)tid) * 6u;
        orow[0] = o0; orow[1] = o1; orow[2] = o2; orow[3] = o3; orow[4] = o4; orow[5] = 0.0f;
    }
}

extern "C" void kernel_launch(void* const* d_in, const int* in_sizes, int n_in,
                              void* d_out, int out_size, void* d_ws, size_t ws_size,
                              hipStream_t stream) {
    (void)d_ws; (void)ws_size; (void)n_in;
    const float* pred = (const float*)d_in[0];
    float* out = (float*)d_out;
    int B = out_size / (MAXDET * 6);           // 64
    int N = in_sizes[0] / (6 * B);             // 25200
    (void)hipFuncSetAttribute((const void*)yolo_nms_kernel,
                              hipFuncAttributeMaxDynamicSharedMemorySize,
                              (int)SMEM_BYTES);
    yolo_nms_kernel<<<B, 1024, SMEM_BYTES, stream>>>(pred, out, N);
}